// MSRPLoss_25812753449609
// MI455X (gfx1250) — compile-verified
//
#include <hip/hip_runtime.h>
#include <hip/hip_bf16.h>
#include <stdint.h>

// Problem constants (from reference setup_inputs)
#define B_ROWS 65536
#define C_COLS 1000
#define ROW_BYTES_REAL 4000   // 1000 f32
#define ROW_BUF 4096          // padded per-row LDS buffer (16B aligned)
#define WAVES_PER_BLOCK 4
#define THREADS_MAIN (WAVES_PER_BLOCK * 32)
#define NBLOCKS 4096

// Issue one row (4000B) as 8 async global->LDS b128 copies.
// 512B per instruction (32 lanes x 16B), last instruction masked to 26 lanes
// (4000 = 7*512 + 26*16). Always 8 async ops per wave per row.
__device__ __forceinline__ void issue_row_copy(const float* __restrict__ row_ptr,
                                               uint32_t lds_base, int lane) {
    const char* g = (const char*)row_ptr;
#pragma unroll
    for (int k = 0; k < 7; ++k) {
        uint64_t ga = (uint64_t)(uintptr_t)(g + (k * 512 + lane * 16));
        uint32_t la = lds_base + (uint32_t)(k * 512 + lane * 16);
        asm volatile("global_load_async_to_lds_b128 %0, %1, off"
                     :: "v"(la), "v"(ga) : "memory");
    }
    if (lane < 26) {
        uint64_t ga = (uint64_t)(uintptr_t)(g + (7 * 512 + lane * 16));
        uint32_t la = lds_base + (uint32_t)(7 * 512 + lane * 16);
        asm volatile("global_load_async_to_lds_b128 %0, %1, off"
                     :: "v"(la), "v"(ga) : "memory");
    }
}

__global__ __launch_bounds__(THREADS_MAIN)
void msrp_main(const float* __restrict__ outputs,
               const int*   __restrict__ labels,
               float* __restrict__ ws_loss,
               int*   __restrict__ ws_cnt) {
    __shared__ __align__(16) char smem[WAVES_PER_BLOCK * 2 * ROW_BUF];
    __shared__ float red_loss[WAVES_PER_BLOCK];
    __shared__ int   red_cnt[WAVES_PER_BLOCK];

    const int lane   = (int)(threadIdx.x & 31u);
    const int wave   = (int)(threadIdx.x >> 5u);
    const int gwave  = (int)blockIdx.x * WAVES_PER_BLOCK + wave;
    const int nwaves = (int)gridDim.x * WAVES_PER_BLOCK;

    // Low 32 bits of the generic shared-pointer == LDS byte offset
    // (AS3 -> flat lowering places the LDS offset in addr[31:0]).
    const uint32_t lds_wave_base =
        (uint32_t)(uintptr_t)(void*)smem + (uint32_t)wave * 2u * ROW_BUF;

    float loss_acc = 0.0f;
    int   cnt_acc  = 0;

    int r = gwave;
    if (r < B_ROWS) {
        uint32_t parity = 0;
        issue_row_copy(outputs + (size_t)r * C_COLS, lds_wave_base, lane);
        for (;;) {
            const int  rn   = r + nwaves;
            const bool more = (rn < B_ROWS);
            if (more) {
                issue_row_copy(outputs + (size_t)rn * C_COLS,
                               lds_wave_base + (parity ^ 1u) * ROW_BUF, lane);
                // Current row (oldest 8 async ops) must be complete; async
                // loads retire in order, so <=8 outstanding == row r landed.
                asm volatile("s_wait_asynccnt 8" ::: "memory");
            } else {
                asm volatile("s_wait_asynccnt 0" ::: "memory");
            }

            const float*  rowf = (const float*)(smem + (size_t)wave * 2 * ROW_BUF
                                                     + (size_t)parity * ROW_BUF);
            const float4* rowv = (const float4*)rowf;

            const int   lab = labels[r];
            const float g   = rowf[lab];   // uniform LDS broadcast

            float hsum = 0.0f;
            float vmax = -3.402823466e+38f;
            int   imax = 0;
            // 250 float4 per row: lanes 0..25 run 8 iters, 26..31 run 7.
            for (int j = lane; j < (C_COLS / 4); j += 32) {
                const float4 x = rowv[j];
                const int base = j * 4;
                hsum += fmaxf(x.x - g + 1.0f, 0.0f);
                hsum += fmaxf(x.y - g + 1.0f, 0.0f);
                hsum += fmaxf(x.z - g + 1.0f, 0.0f);
                hsum += fmaxf(x.w - g + 1.0f, 0.0f);
                if (x.x > vmax) { vmax = x.x; imax = base;     }
                if (x.y > vmax) { vmax = x.y; imax = base + 1; }
                if (x.z > vmax) { vmax = x.z; imax = base + 2; }
                if (x.w > vmax) { vmax = x.w; imax = base + 3; }
            }
            // wave32 butterfly reduction (deterministic, first-occurrence ties)
#pragma unroll
            for (int off = 16; off > 0; off >>= 1) {
                const float h2 = __shfl_xor(hsum, off, 32);
                const float v2 = __shfl_xor(vmax, off, 32);
                const int   i2 = __shfl_xor(imax, off, 32);
                hsum += h2;
                if (v2 > vmax || (v2 == vmax && i2 < imax)) { vmax = v2; imax = i2; }
            }
            loss_acc += hsum - 1.0f;          // remove j==label term (exactly 1)
            cnt_acc  += (imax == lab) ? 1 : 0;

            if (!more) break;
            r = rn;
            parity ^= 1u;
        }
    }

    if (lane == 0) { red_loss[wave] = loss_acc; red_cnt[wave] = cnt_acc; }
    __syncthreads();
    if (threadIdx.x == 0) {
        float l = 0.0f; int c = 0;
#pragma unroll
        for (int w = 0; w < WAVES_PER_BLOCK; ++w) { l += red_loss[w]; c += red_cnt[w]; }
        ws_loss[blockIdx.x] = l;
        ws_cnt[blockIdx.x]  = c;
    }
}

__global__ __launch_bounds__(256)
void msrp_reduce(const float* __restrict__ ws_loss,
                 const int*   __restrict__ ws_cnt,
                 float* __restrict__ out, int nparts) {
    __shared__ float sl[256];
    __shared__ int   sc[256];
    const int t = (int)threadIdx.x;
    float l = 0.0f; int c = 0;
    for (int i = t; i < nparts; i += 256) { l += ws_loss[i]; c += ws_cnt[i]; }
    sl[t] = l; sc[t] = c;
    __syncthreads();
    for (int s = 128; s > 0; s >>= 1) {
        if (t < s) { sl[t] += sl[t + s]; sc[t] += sc[t + s]; }
        __syncthreads();
    }
    if (t == 0) {
        out[0] = sl[0] * (1.0f / (float)B_ROWS);  // mean loss
        out[1] = (float)sc[0];                    // correct count (exact in f32)
    }
}

extern "C" void kernel_launch(void* const* d_in, const int* in_sizes, int n_in,
                              void* d_out, int out_size, void* d_ws, size_t ws_size,
                              hipStream_t stream) {
    (void)in_sizes; (void)n_in; (void)out_size; (void)ws_size;
    const float* outputs = (const float*)d_in[0];
    const int*   labels  = (const int*)d_in[1];
    float* out     = (float*)d_out;
    float* ws_loss = (float*)d_ws;
    int*   ws_cnt  = (int*)((char*)d_ws + NBLOCKS * sizeof(float));

    msrp_main<<<NBLOCKS, THREADS_MAIN, 0, stream>>>(outputs, labels, ws_loss, ws_cnt);
    msrp_reduce<<<1, 256, 0, stream>>>(ws_loss, ws_cnt, out, NBLOCKS);
}